// Graph_55843164783140
// MI455X (gfx1250) — compile-verified
//
#include <hip/hip_runtime.h>
#include <math.h>

#define N_NODES   50000
#define N_EDGES   1600000
#define HIDDEN    64
#define TWOH      128
#define SLOPE     0.2f
#define ROWSTRIDE 132   // 128 + 4 pad floats: kills LDS bank conflicts, keeps 16B align

typedef __attribute__((ext_vector_type(2))) float v2f;
typedef __attribute__((ext_vector_type(8))) float v8f;

// Order-preserving float<->uint encoding so we can use integer atomicMax for
// the segmented max (guaranteed single-instruction HW atomic path).
__device__ __forceinline__ unsigned encf(float f) {
  unsigned u = __float_as_uint(f);
  return ((int)u < 0) ? ~u : (u | 0x80000000u);
}
__device__ __forceinline__ float decf(unsigned u) {
  return __uint_as_float(((int)u < 0) ? (u ^ 0x80000000u) : ~u);
}

// ---- init: zero encoded-segmax (0 == below any real encoding), denom, numer
__global__ void init_kernel(unsigned* __restrict__ segmax,
                            float* __restrict__ denom,
                            float* __restrict__ numer, int n) {
  int i = blockIdx.x * blockDim.x + threadIdx.x;
  if (i < n) { segmax[i] = 0u; denom[i] = 0.0f; numer[i] = 0.0f; }
}

// ---- pass 1: alpha = leaky_relu(concat(state[src],state[dst]) @ W) * ef
//      via V_WMMA_F32_16X16X4_F32; also atomic segment-max into segmax[dst].
// Block = 256 threads = 8 waves. Block handles 32 edges per tile iteration:
//   wave w -> edge subtile (w>>2)*16, output cols ((w&3)*16 .. +16).
// Weight slice for a wave's 16 columns (128x16 f32) lives in 64 VGPRs.
__launch_bounds__(256)
__global__ void pass1_kernel(const float* __restrict__ state,
                             const float* __restrict__ W,
                             const float* __restrict__ ef,
                             const int* __restrict__ src,
                             const int* __restrict__ dst,
                             float* __restrict__ alpha,
                             unsigned* __restrict__ segmax,
                             int nTiles) {
  __shared__ float tile[32 * ROWSTRIDE];
  __shared__ int   dstL[32];
  __shared__ float efL[32];

  const int tid  = threadIdx.x;
  const int lane = tid & 31;
  const int wave = tid >> 5;
  const int hh   = lane >> 4;        // lane half (0/1)
  const int nn   = lane & 15;        // position within half
  const int st   = wave >> 2;        // edge subtile (0/1)
  const int nb   = (wave & 3) << 4;  // output column base

  // Load this wave's B fragments once: bf[kk] covers K = 4*kk .. 4*kk+3.
  // B 4x16 layout: VGPR0 = {K0 | K2}, VGPR1 = {K1 | K3} across lane halves.
  v2f bf[32];
#pragma unroll
  for (int kk = 0; kk < 32; ++kk) {
    int k = 4 * kk + 2 * hh;
    bf[kk].x = W[(size_t)k * HIDDEN + nb + nn];
    bf[kk].y = W[(size_t)(k + 1) * HIDDEN + nb + nn];
  }

  for (int t = blockIdx.x; t < nTiles; t += gridDim.x) {
    const int eBase = t * 32;

    // Cooperative gather: 32 edges x 128 comps -> LDS (16 KB).
    // thread -> (edge = tid/8, part = tid%8); part<4: src comps, else dst comps.
    {
      const int le   = tid >> 3;
      const int part = tid & 7;
      const int e    = eBase + le;
      const int idx  = (part < 4) ? src[e] : dst[e];
      const int comp = (part & 3) * 16;
      const float* g = state + (size_t)idx * HIDDEN + comp;
      float* l = tile + le * ROWSTRIDE + ((part < 4) ? comp : (64 + comp));
#pragma unroll
      for (int j = 0; j < 4; ++j)
        *(float4*)(l + 4 * j) = *(const float4*)(g + 4 * j);
      if (tid < 32) {
        dstL[tid] = dst[eBase + tid];
        efL[tid]  = ef[eBase + tid];
      }
    }
    __syncthreads();

    // A 16x4 layout: lanes 0-15 hold M rows; VGPR0 = {K0 | K2}, VGPR1 = {K1 | K3}.
    const float* arow = tile + (st * 16 + nn) * ROWSTRIDE;
    v8f acc = {0.f, 0.f, 0.f, 0.f, 0.f, 0.f, 0.f, 0.f};
#pragma unroll
    for (int kk = 0; kk < 32; ++kk) {
      const int koff = 4 * kk + 2 * hh;
      v2f a;
      a.x = arow[koff];
      a.y = arow[koff + 1];
      acc = __builtin_amdgcn_wmma_f32_16x16x4_f32(
          /*neg_a=*/false, a, /*neg_b=*/false, bf[kk],
          /*c_mod=*/(short)0, acc, /*reuse_a=*/false, /*reuse_b=*/false);
    }

    // D layout: VGPR i -> lanes0-15 M=i, lanes16-31 M=i+8; N = nb + nn.
#pragma unroll
    for (int i = 0; i < 8; ++i) {
      const int r   = st * 16 + i + (hh ? 8 : 0);   // local edge row 0..31
      const int col = nb + nn;
      float x = acc[i];
      x = (x > 0.0f) ? x : SLOPE * x;               // leaky_relu
      x *= efL[r];                                  // edge_feature scale
      const int e = eBase + r;
      __builtin_nontemporal_store(x, alpha + (size_t)e * HIDDEN + col);
      atomicMax(segmax + (size_t)dstL[r] * HIDDEN + col, encf(x));
    }
    __syncthreads();
  }
}

// ---- pass 2: ex = exp(alpha - segmax[dst]); accumulate denom and ex*state[src]
__launch_bounds__(256)
__global__ void pass2_kernel(const float* __restrict__ state,
                             const int* __restrict__ src,
                             const int* __restrict__ dst,
                             const float* __restrict__ alpha,
                             const unsigned* __restrict__ segmax,
                             float* __restrict__ denom,
                             float* __restrict__ numer) {
  const size_t t = (size_t)blockIdx.x * blockDim.x + threadIdx.x;
  if (t >= (size_t)N_EDGES * HIDDEN) return;
  const int e = (int)(t >> 6);
  const int h = (int)(t & 63);
  const int d = dst[e];
  const int s = src[e];
  const float a  = __builtin_nontemporal_load(alpha + t);
  const float mx = decf(segmax[(size_t)d * HIDDEN + h]);
  const float ex = __expf(a - mx);
  unsafeAtomicAdd(denom + (size_t)d * HIDDEN + h, ex);
  unsafeAtomicAdd(numer + (size_t)d * HIDDEN + h,
                  ex * state[(size_t)s * HIDDEN + h]);
}

// ---- pass 3: out = relu(numer / denom), 0 for zero-in-degree nodes
__global__ void pass3_kernel(const float* __restrict__ numer,
                             const float* __restrict__ denom,
                             float* __restrict__ out, int n) {
  int i = blockIdx.x * blockDim.x + threadIdx.x;
  if (i >= n) return;
  const float q = denom[i];
  const float v = (q != 0.0f) ? (numer[i] / q) : 0.0f;
  out[i] = fmaxf(v, 0.0f);
}

extern "C" void kernel_launch(void* const* d_in, const int* in_sizes, int n_in,
                              void* d_out, int out_size, void* d_ws, size_t ws_size,
                              hipStream_t stream) {
  const float* state  = (const float*)d_in[0];
  // d_in[1] = feature: unused by reference
  const float* weight = (const float*)d_in[2];
  const float* ef     = (const float*)d_in[3];
  const int*   src    = (const int*)d_in[4];
  const int*   dst    = (const int*)d_in[5];
  float* out = (float*)d_out;

  char* ws = (char*)d_ws;
  const size_t alphaBytes = (size_t)N_EDGES * HIDDEN * sizeof(float);
  const size_t nodeBytes  = (size_t)N_NODES * HIDDEN * sizeof(float);
  float*    alpha  = (float*)ws;
  unsigned* segmax = (unsigned*)(ws + alphaBytes);
  float*    denom  = (float*)(ws + alphaBytes + nodeBytes);
  float*    numer  = (float*)(ws + alphaBytes + 2 * nodeBytes);

  const int nNode = N_NODES * HIDDEN;
  init_kernel<<<(nNode + 255) / 256, 256, 0, stream>>>(segmax, denom, numer, nNode);

  const int nTiles = N_EDGES / 32;  // 50000, exact
  pass1_kernel<<<2048, 256, 0, stream>>>(state, weight, ef, src, dst,
                                         alpha, segmax, nTiles);

  const size_t tot = (size_t)N_EDGES * HIDDEN;
  pass2_kernel<<<(unsigned)((tot + 255) / 256), 256, 0, stream>>>(
      state, src, dst, alpha, segmax, denom, numer);

  pass3_kernel<<<(nNode + 255) / 256, 256, 0, stream>>>(numer, denom, out, nNode);
}